// STJGAT_23450521436522
// MI455X (gfx1250) — compile-verified
//
#include <hip/hip_runtime.h>
#include <hip/hip_bf16.h>

typedef __attribute__((ext_vector_type(2))) float v2f;
typedef __attribute__((ext_vector_type(8))) float v8f;

typedef __attribute__((ext_vector_type(4))) unsigned int tdm_u32x4;
typedef __attribute__((ext_vector_type(8))) int tdm_i32x8;
typedef __attribute__((ext_vector_type(4))) int tdm_i32x4;

#define EPSF 1e-5f
#define SELU_SCALE 1.0507009873554805f
#define SELU_ALPHA 1.6732632423543772f

// LDS row stride (floats) for the TDM-staged h block: 32 data + 2 pad DWORDs.
// 34*lr mod 64 is distinct across lanes -> conflict-free ds reads, 8B aligned.
#define HPAD 34

__device__ __forceinline__ float selu_f(float x) {
    return SELU_SCALE * (x > 0.f ? x : SELU_ALPHA * (expf(x) - 1.f));
}

// ---------------------------------------------------------------------------
// TDM: DMA a 2D tile (dim1 rows x dim0 f32 elems, contiguous rows) from global
// into LDS at byte offset lds_off, inserting 2 pad DWORDs after every 32
// DWORDs (pad_interval=4 -> 32 DWORDs, pad_amount=1 -> 2 DWORDs) so the LDS
// row stride becomes HPAD floats.  D# per CDNA5 ISA ch.8 (groups 0/1; groups
// 2/3 NULL-equivalent zeros for a 2D tensor).
// ---------------------------------------------------------------------------
__device__ __forceinline__ void tdm_load_2d_to_lds(unsigned lds_off,
                                                   const void* gptr,
                                                   unsigned dim0, unsigned dim1) {
    const unsigned long long ga = (unsigned long long)gptr;
    tdm_u32x4 g0;
    g0[0] = 1u;                                            // count=1 (valid user D#)
    g0[1] = lds_off;                                       // lds_addr (bytes)
    g0[2] = (unsigned)(ga & 0xFFFFFFFFu);                  // global_addr[31:0]
    g0[3] = (unsigned)((ga >> 32) & 0x1FFFFFFu) | (2u << 30); // addr[56:32] | type=2
    tdm_i32x8 g1;
    g1[0] = (2 << 16) | (1 << 20) | (4 << 22) | (1 << 25); // data_size=4B, pad_en,
                                                           // pad_interval=32dw, pad=2dw
    g1[1] = (int)((dim0 & 0xFFFFu) << 16);                 // tensor_dim0[15:0] @ [31:16]
    g1[2] = (int)(((dim0 >> 16) & 0xFFFFu) |               // tensor_dim0[31:16]
                  ((dim1 & 0xFFFFu) << 16));               // tensor_dim1[15:0]
    g1[3] = (int)(((dim1 >> 16) & 0xFFFFu) |               // tensor_dim1[31:16]
                  ((dim0 & 0xFFFFu) << 16));               // tile_dim0
    g1[4] = (int)(dim1 & 0xFFFFu);                         // tile_dim1 (tile_dim2=0)
    g1[5] = (int)dim0;                                     // tensor_dim0_stride[31:0]
    g1[6] = 0;                                             // stride hi / dim1_stride lo
    g1[7] = 0;                                             // dim1_stride hi
    const tdm_i32x4 z4 = {0, 0, 0, 0};
#if __has_include(<hip/amd_detail/amd_gfx1250_TDM.h>)
    const tdm_i32x8 z8 = {0, 0, 0, 0, 0, 0, 0, 0};
    __builtin_amdgcn_tensor_load_to_lds(g0, g1, z4, z4, z8, 0);
#else
    __builtin_amdgcn_tensor_load_to_lds(g0, g1, z4, z4, 0);
#endif
}

// ---------------------------------------------------------------------------
// Kernel 1: fused conv1(32x32 GEMM via v_wmma_f32_16x16x4_f32) + SELU + BN +
// conv2 (dot over 32 channels) -> m[b,t,f].  One wave handles 16 rows; block
// of 256 threads (8 waves) handles 128 rows; grid = 524288/128 = 4096.
// ---------------------------------------------------------------------------
__global__ void k_conv_bn_score(const float* __restrict__ x,
                                const float* __restrict__ c1w,
                                const float* __restrict__ c1b,
                                const float* __restrict__ bng,
                                const float* __restrict__ bnb,
                                const float* __restrict__ bnm,
                                const float* __restrict__ bnv,
                                const float* __restrict__ c2w,
                                const float* __restrict__ c2b,
                                float* __restrict__ mout) {
    const int lane = threadIdx.x & 31;
    const int wave = threadIdx.x >> 5;
    const int hi = lane >> 4;          // half-wave select (K phase)
    const int lr = lane & 15;          // row (A) / col (B,C)
    const int rowbase = blockIdx.x * 128 + wave * 16;
    const long arow = rowbase + lr;

    v2f bt0[8], bt1[8], a2[8];
#pragma unroll
    for (int s = 0; s < 8; ++s) {
        const int k = 4 * s + 2 * hi;
        // B[k][n] = c1w[n*32+k]  (out[r][o] = sum_c A[r][c]*c1w[o][c])
        bt0[s] = *(const v2f*)(c1w + lr * 32 + k);
        bt1[s] = *(const v2f*)(c1w + (lr + 16) * 32 + k);
        a2[s]  = *(const v2f*)(x + arow * 32 + k);
    }
    v8f acc0 = {}; v8f acc1 = {};
#pragma unroll
    for (int s = 0; s < 8; ++s) {
        acc0 = __builtin_amdgcn_wmma_f32_16x16x4_f32(false, a2[s], false, bt0[s],
                                                     (short)0, acc0, false, false);
        acc1 = __builtin_amdgcn_wmma_f32_16x16x4_f32(false, a2[s], false, bt1[s],
                                                     (short)0, acc1, false, false);
    }
    const int o0 = lr, o1 = lr + 16;
    const float sc0 = bng[o0] * rsqrtf(bnv[o0] + EPSF);
    const float of0 = bnb[o0] - bnm[o0] * sc0;
    const float sc1 = bng[o1] * rsqrtf(bnv[o1] + EPSF);
    const float of1 = bnb[o1] - bnm[o1] * sc1;
    const float cb0 = c1b[o0], cb1 = c1b[o1];
    const float w0 = c2w[o0], w1v = c2w[o1];
    const float c2 = c2b[0];
#pragma unroll
    for (int v = 0; v < 8; ++v) {
        const float val0 = selu_f(acc0[v] + cb0) * sc0 + of0;
        const float val1 = selu_f(acc1[v] + cb1) * sc1 + of1;
        float p = val0 * w0 + val1 * w1v;
        p += __shfl_xor(p, 1, 32);
        p += __shfl_xor(p, 2, 32);
        p += __shfl_xor(p, 4, 32);
        p += __shfl_xor(p, 8, 32);
        if (lr == 0) mout[rowbase + v + 8 * hi] = p + c2;
    }
}

// ---------------------------------------------------------------------------
// Kernel 2: softmax over T (axis=1 of (B,T,F)).  One block per (b,f) column.
// ---------------------------------------------------------------------------
__global__ void k_softmax_T(float* __restrict__ m) {
    const int bf = blockIdx.x;
    const int b = bf >> 7;
    const int f = bf & 127;
    const int t = threadIdx.x;              // blockDim = 256 = T
    __shared__ float red[256];
    const int idx = (b * 256 + t) * 128 + f;
    const float v = m[idx];
    red[t] = v; __syncthreads();
    for (int s = 128; s > 0; s >>= 1) {
        if (t < s) red[t] = fmaxf(red[t], red[t + s]);
        __syncthreads();
    }
    const float mx = red[0]; __syncthreads();
    const float e = expf(v - mx);
    red[t] = e; __syncthreads();
    for (int s = 128; s > 0; s >>= 1) {
        if (t < s) red[t] += red[t + s];
        __syncthreads();
    }
    m[idx] = e / red[0];
}

// ---------------------------------------------------------------------------
// Kernel 3a/3b: attention pools  FCatt[b,f,c] = sum_t M0*x,  TCatt[b,t,c] = sum_f M0*x
// ---------------------------------------------------------------------------
__global__ void k_fcatt(const float* __restrict__ x, const float* __restrict__ M0,
                        float* __restrict__ FCatt) {
    const int bf = blockIdx.x;
    const int b = bf >> 7, f = bf & 127;
    const int c = threadIdx.x;              // 32 threads
    float acc = 0.f;
    for (int t = 0; t < 256; ++t) {
        const int idx = (b * 256 + t) * 128 + f;
        __builtin_prefetch(x + (long)idx * 32 + 8 * 128 * 32, 0, 0);
        acc += M0[idx] * x[(long)idx * 32 + c];
    }
    FCatt[bf * 32 + c] = acc;
}

__global__ void k_tcatt(const float* __restrict__ x, const float* __restrict__ M0,
                        float* __restrict__ TCatt) {
    const int bt = blockIdx.x;              // b*256+t
    const int c = threadIdx.x;              // 32 threads
    const int base = bt * 128;
    float acc = 0.f;
    for (int f = 0; f < 128; ++f) {
        const int idx = base + f;
        __builtin_prefetch(x + (long)idx * 32 + 8 * 32, 0, 0);
        acc += M0[idx] * x[(long)idx * 32 + c];
    }
    TCatt[bt * 32 + c] = acc;
}

// ---------------------------------------------------------------------------
// Kernel 4: GAT block.  One 256-thread block per (b,i).  h_b (N x 32 f32) is
// DMA-staged into LDS by the Tensor Data Mover (padded to stride HPAD for
// conflict-free reads), then the (N x 32)@(32 x 32) GEMM
// E = tanh((h_i (*) h_j) @ apw^T) runs on v_wmma_f32_16x16x4_f32; per-row
// score reduction via shuffles, softmax over j in LDS, tiny per-row output.
// ---------------------------------------------------------------------------
__global__ void k_gat(const float* __restrict__ h, int N,
                      const float* __restrict__ apw, const float* __restrict__ apb,
                      const float* __restrict__ aw,
                      const float* __restrict__ pww, const float* __restrict__ pwb,
                      const float* __restrict__ poww, const float* __restrict__ pob,
                      const float* __restrict__ gg, const float* __restrict__ be,
                      const float* __restrict__ mu, const float* __restrict__ var,
                      float* __restrict__ out) {
    const int tid = threadIdx.x;
    const int lane = tid & 31, wave = tid >> 5;
    const int hi = lane >> 4, lr = lane & 15;
    const int b = blockIdx.x / N;
    const int i = blockIdx.x % N;
    __shared__ float s_h[256 * HPAD];       // TDM-staged h_b, padded rows
    __shared__ float s_sc[256];
    __shared__ float red[256];
    __shared__ float s_part[8][32];
    __shared__ float s_row[32];
    const float* hb = h + (long)b * N * 32;

    // One wave issues the tensor DMA for the whole h_b block, waits on
    // TENSORcnt, then the workgroup barrier publishes LDS to all waves.
    if (wave == 0) {
        tdm_load_2d_to_lds((unsigned)(unsigned long long)(const void*)s_h,
                           hb, 32u, (unsigned)N);
        __builtin_amdgcn_s_wait_tensorcnt(0);
    }
    __syncthreads();

    const float* hi_row = s_h + i * HPAD;   // h_i from the staged block

    v2f bt0[8], bt1[8];
#pragma unroll
    for (int s = 0; s < 8; ++s) {
        const int k = 4 * s + 2 * hi;
        bt0[s] = *(const v2f*)(apw + lr * 32 + k);
        bt1[s] = *(const v2f*)(apw + (lr + 16) * 32 + k);
    }
    const int tilesPerWave = N >> 7;        // N/128 : 1 or 2
    const float aw0 = aw[lr], aw1 = aw[lr + 16];
    const float ab0 = apb[lr], ab1 = apb[lr + 16];
    for (int tt = 0; tt < tilesPerWave; ++tt) {
        const int jbase = (wave * tilesPerWave + tt) * 16;
        const int j = jbase + lr;
        v2f a2[8];
#pragma unroll
        for (int s = 0; s < 8; ++s) {
            const int k = 4 * s + 2 * hi;
            v2f hv = *(const v2f*)(s_h + j * HPAD + k);   // 8B aligned (HPAD even)
            a2[s].x = hv.x * hi_row[k];
            a2[s].y = hv.y * hi_row[k + 1];
        }
        v8f acc0 = {}; v8f acc1 = {};
#pragma unroll
        for (int s = 0; s < 8; ++s) {
            acc0 = __builtin_amdgcn_wmma_f32_16x16x4_f32(false, a2[s], false, bt0[s],
                                                         (short)0, acc0, false, false);
            acc1 = __builtin_amdgcn_wmma_f32_16x16x4_f32(false, a2[s], false, bt1[s],
                                                         (short)0, acc1, false, false);
        }
#pragma unroll
        for (int v = 0; v < 8; ++v) {
            const float e0 = tanhf(acc0[v] + ab0);
            const float e1 = tanhf(acc1[v] + ab1);
            float p = e0 * aw0 + e1 * aw1;
            p += __shfl_xor(p, 1, 32);
            p += __shfl_xor(p, 2, 32);
            p += __shfl_xor(p, 4, 32);
            p += __shfl_xor(p, 8, 32);
            if (lr == 0) s_sc[jbase + v + 8 * hi] = p;
        }
    }
    __syncthreads();

    // softmax over j = 0..N-1
    const float myv = (tid < N) ? s_sc[tid] : -3.4e38f;
    red[tid] = myv; __syncthreads();
    for (int s = 128; s > 0; s >>= 1) {
        if (tid < s) red[tid] = fmaxf(red[tid], red[tid + s]);
        __syncthreads();
    }
    const float mx = red[0]; __syncthreads();
    const float e = (tid < N) ? expf(myv - mx) : 0.f;
    red[tid] = e; __syncthreads();
    for (int s = 128; s > 0; s >>= 1) {
        if (tid < s) red[tid] += red[tid + s];
        __syncthreads();
    }
    const float sum = red[0]; __syncthreads();
    if (tid < N) s_sc[tid] = e / sum;
    __syncthreads();

    // attrow[c] = sum_j att[j] * h[b,j,c]   (h rows from LDS)
    {
        const int g = tid >> 5, c = tid & 31;
        float p = 0.f;
        for (int j = g; j < N; j += 8) p += s_sc[j] * s_h[j * HPAD + c];
        s_part[g][c] = p;
    }
    __syncthreads();
    if (tid < 32) {
        float p = 0.f;
#pragma unroll
        for (int g = 0; g < 8; ++g) p += s_part[g][tid];
        s_row[tid] = p;
    }
    __syncthreads();
    if (tid < 32) {
        const int d = tid;
        float o = pwb[d] + pob[d];
#pragma unroll
        for (int c = 0; c < 32; ++c)
            o += s_row[c] * pww[d * 32 + c] + hi_row[c] * poww[d * 32 + c];
        o = (o - mu[d]) * rsqrtf(var[d] + EPSF) * gg[d] + be[d];
        out[((long)b * N + i) * 32 + d] = selu_f(o);
    }
}

// ---------------------------------------------------------------------------
// Kernel 5: score + top-k pooling (stable descending rank-select; matches
// lax.top_k ordering).  One block per b.
// ---------------------------------------------------------------------------
__global__ void k_pool(const float* __restrict__ h, int N, int kn,
                       const float* __restrict__ pw, const float* __restrict__ pb,
                       float* __restrict__ out) {
    const int b = blockIdx.x;
    const int tid = threadIdx.x;            // blockDim = 256
    __shared__ float sc[256];
    const float* hb = h + (long)b * N * 32;
    if (tid < N) {
        float s = pb[0];
#pragma unroll
        for (int d = 0; d < 32; ++d) s += hb[tid * 32 + d] * pw[d];
        sc[tid] = 1.f / (1.f + expf(-s));
    }
    __syncthreads();
    if (tid < N) {
        const float myv = sc[tid];
        int rank = 0;
        for (int j = 0; j < N; ++j) {
            const float o = sc[j];
            rank += (o > myv) || (o == myv && j < tid);
        }
        if (rank < kn) {
#pragma unroll
            for (int d = 0; d < 32; ++d)
                out[((long)b * kn + rank) * 32 + d] = hb[tid * 32 + d] * myv;
        }
    }
}

// ---------------------------------------------------------------------------
// Kernel 6: cross-attention fusion + max pool + final FC.  One block per b.
// ---------------------------------------------------------------------------
__global__ void k_final(const float* __restrict__ xF, const float* __restrict__ xT,
                        const float* __restrict__ w1, const float* __restrict__ w2,
                        const float* __restrict__ fcw, const float* __restrict__ fcb,
                        float* __restrict__ outp) {
    const int b = blockIdx.x;
    const int tid = threadIdx.x;            // blockDim = 128
    __shared__ float xFs[64 * 32];
    __shared__ float xTs[128 * 32];
    __shared__ float fF[64 * 32];
    __shared__ float fT[128 * 32];
    __shared__ float node[64];
    for (int idx = tid; idx < 64 * 32; idx += 128) xFs[idx] = xF[(long)b * 64 * 32 + idx];
    for (int idx = tid; idx < 128 * 32; idx += 128) xTs[idx] = xT[(long)b * 128 * 32 + idx];
    __syncthreads();

    if (tid < 64) {                         // xF_fused row tid
        float y[32];
#pragma unroll
        for (int d = 0; d < 32; ++d) {
            float a = 0.f;
            for (int c = 0; c < 32; ++c) a += xFs[tid * 32 + c] * w1[d * 32 + c];
            y[d] = a;
        }
        float scv[128];
        float mx = -3.4e38f;
        for (int j = 0; j < 128; ++j) {
            float a = 0.f;
#pragma unroll
            for (int d = 0; d < 32; ++d) a += y[d] * xTs[j * 32 + d];
            scv[j] = a; mx = fmaxf(mx, a);
        }
        float sum = 0.f;
        for (int j = 0; j < 128; ++j) { scv[j] = expf(scv[j] - mx); sum += scv[j]; }
        const float inv = 1.f / sum;
#pragma unroll
        for (int d = 0; d < 32; ++d) {
            float a = xFs[tid * 32 + d];
            for (int j = 0; j < 128; ++j) a += scv[j] * inv * xTs[j * 32 + d];
            fF[tid * 32 + d] = a;
        }
    }
    {                                       // xT_fused row tid (all 128 threads)
        float y[32];
#pragma unroll
        for (int d = 0; d < 32; ++d) {
            float a = 0.f;
            for (int c = 0; c < 32; ++c) a += xTs[tid * 32 + c] * w2[d * 32 + c];
            y[d] = a;
        }
        float scv[64];
        float mx = -3.4e38f;
        for (int j = 0; j < 64; ++j) {
            float a = 0.f;
#pragma unroll
            for (int d = 0; d < 32; ++d) a += y[d] * xFs[j * 32 + d];
            scv[j] = a; mx = fmaxf(mx, a);
        }
        float sum = 0.f;
        for (int j = 0; j < 64; ++j) { scv[j] = expf(scv[j] - mx); sum += scv[j]; }
        const float inv = 1.f / sum;
#pragma unroll
        for (int d = 0; d < 32; ++d) {
            float a = xTs[tid * 32 + d];
            for (int j = 0; j < 64; ++j) a += scv[j] * inv * xFs[j * 32 + d];
            fT[tid * 32 + d] = a;
        }
    }
    __syncthreads();
    if (tid < 32) {
        float mF = -3.4e38f;
        for (int r = 0; r < 64; ++r) mF = fmaxf(mF, fF[r * 32 + tid]);
        node[tid] = mF;
    } else if (tid < 64) {
        const int d = tid - 32;
        float mT = -3.4e38f;
        for (int r = 0; r < 128; ++r) mT = fmaxf(mT, fT[r * 32 + d]);
        node[tid] = mT;
    }
    __syncthreads();
    if (tid < 2) {
        float a = fcb[tid];
        for (int k2 = 0; k2 < 64; ++k2) a += node[k2] * fcw[tid * 64 + k2];
        outp[b * 2 + tid] = a;
    }
}

// ---------------------------------------------------------------------------
extern "C" void kernel_launch(void* const* d_in, const int* in_sizes, int n_in,
                              void* d_out, int out_size, void* d_ws, size_t ws_size,
                              hipStream_t stream) {
    const float* x    = (const float*)d_in[0];
    const float* c1w  = (const float*)d_in[1];
    const float* c1b  = (const float*)d_in[2];
    const float* bng  = (const float*)d_in[3];
    const float* bnb  = (const float*)d_in[4];
    const float* bnm  = (const float*)d_in[5];
    const float* bnv  = (const float*)d_in[6];
    const float* c2w  = (const float*)d_in[7];
    const float* c2b  = (const float*)d_in[8];
    const float* fc_apw = (const float*)d_in[9];
    const float* fc_apb = (const float*)d_in[10];
    const float* fc_aw  = (const float*)d_in[11];
    const float* fc_pww = (const float*)d_in[12];
    const float* fc_pwb = (const float*)d_in[13];
    const float* fc_pow = (const float*)d_in[14];
    const float* fc_pob = (const float*)d_in[15];
    const float* fc_g   = (const float*)d_in[16];
    const float* fc_be  = (const float*)d_in[17];
    const float* fc_m   = (const float*)d_in[18];
    const float* fc_v   = (const float*)d_in[19];
    const float* tc_apw = (const float*)d_in[20];
    const float* tc_apb = (const float*)d_in[21];
    const float* tc_aw  = (const float*)d_in[22];
    const float* tc_pww = (const float*)d_in[23];
    const float* tc_pwb = (const float*)d_in[24];
    const float* tc_pow = (const float*)d_in[25];
    const float* tc_pob = (const float*)d_in[26];
    const float* tc_g   = (const float*)d_in[27];
    const float* tc_be  = (const float*)d_in[28];
    const float* tc_m   = (const float*)d_in[29];
    const float* tc_v   = (const float*)d_in[30];
    const float* pfw  = (const float*)d_in[31];
    const float* pfb  = (const float*)d_in[32];
    const float* ptw  = (const float*)d_in[33];
    const float* ptb  = (const float*)d_in[34];
    const float* w1   = (const float*)d_in[35];
    const float* w2   = (const float*)d_in[36];
    const float* fcw  = (const float*)d_in[37];
    const float* fcb  = (const float*)d_in[38];

    float* ws    = (float*)d_ws;
    float* m     = ws;                       // B*T*F       = 524288
    float* FCatt = m + 524288;               // B*128*32    = 65536
    float* TCatt = FCatt + 65536;            // B*256*32    = 131072
    float* FCg   = TCatt + 131072;           // 65536
    float* TCg   = FCg + 65536;              // 131072
    float* xF    = TCg + 131072;             // B*64*32     = 32768
    float* xT    = xF + 32768;               // B*128*32    = 65536

    k_conv_bn_score<<<4096, 256, 0, stream>>>(x, c1w, c1b, bng, bnb, bnm, bnv, c2w, c2b, m);
    k_softmax_T<<<2048, 256, 0, stream>>>(m);
    k_fcatt<<<2048, 32, 0, stream>>>(x, m, FCatt);
    k_tcatt<<<4096, 32, 0, stream>>>(x, m, TCatt);
    k_gat<<<16 * 128, 256, 0, stream>>>(FCatt, 128, fc_apw, fc_apb, fc_aw, fc_pww, fc_pwb,
                                        fc_pow, fc_pob, fc_g, fc_be, fc_m, fc_v, FCg);
    k_gat<<<16 * 256, 256, 0, stream>>>(TCatt, 256, tc_apw, tc_apb, tc_aw, tc_pww, tc_pwb,
                                        tc_pow, tc_pob, tc_g, tc_be, tc_m, tc_v, TCg);
    k_pool<<<16, 256, 0, stream>>>(FCg, 128, 64, pfw, pfb, xF);
    k_pool<<<16, 256, 0, stream>>>(TCg, 256, 128, ptw, ptb, xT);
    k_final<<<16, 128, 0, stream>>>(xF, xT, w1, w2, fcw, fcb, (float*)d_out);
}